// MoleGen_31671088841193
// MI455X (gfx1250) — compile-verified
//
#include <hip/hip_runtime.h>

#define EMBD      16
#define N_NODES   32768
#define N_EDGES   65536
#define NGRAPH    1024
#define FC_E      1015808
#define NUM_LAYERS 4

typedef float v2f __attribute__((ext_vector_type(2)));
typedef float v4f __attribute__((ext_vector_type(4)));
typedef float v8f __attribute__((ext_vector_type(8)));

#define V8ZERO {0.f,0.f,0.f,0.f,0.f,0.f,0.f,0.f}

// ---------------------------------------------------------------------------
// WMMA helper: acc += X(16x16) @ W(16x16) via 4x V_WMMA_F32_16X16X4_F32
// A (16x4 f32): lanes 0-15 -> M with K={k0,k0+1}; lanes 16-31 -> K={k0+2,k0+3}
// D: lane n<16: vgpr r = D[r][n]; lane n+16: vgpr r = D[r+8][n]
// ---------------------------------------------------------------------------
__device__ __forceinline__ v8f wmma_tile16(const float* __restrict__ X,
                                           const float* __restrict__ W,
                                           v8f acc) {
  const int lane = threadIdx.x & 31;
  const int nm   = lane & 15;
  const int kh   = (lane >> 4) << 1;
#pragma unroll
  for (int kk = 0; kk < 4; ++kk) {
    const int k0 = kk * 4 + kh;
    v2f a, b;
    const float* xr = X + nm * EMBD + k0;
    a.x = xr[0];
    a.y = xr[1];
    b.x = W[k0 * EMBD + nm];
    b.y = W[(k0 + 1) * EMBD + nm];
    acc = __builtin_amdgcn_wmma_f32_16x16x4_f32(false, a, false, b,
                                                (short)0, acc, false, false);
  }
  return acc;
}

__device__ __forceinline__ void store_tile16(float* __restrict__ Y, v8f c,
                                             const float* __restrict__ bias,
                                             int relu) {
  const int lane = threadIdx.x & 31;
  const int col  = lane & 15;
  const int mh   = (lane >> 4) * 8;
  const float bv = bias ? bias[col] : 0.0f;
#pragma unroll
  for (int r = 0; r < 8; ++r) {
    float v = c[r] + bv;
    if (relu) v = fmaxf(v, 0.0f);
    Y[(mh + r) * EMBD + col] = v;
  }
}

__device__ __forceinline__ float sigm(float x) {
  return 1.0f / (1.0f + __expf(-x));
}

// ---------------------------------------------------------------------------
// Generic 16->16 linear (z-phase hb/hc, mlp l1, u)
// ---------------------------------------------------------------------------
__global__ void k_lin16(const float* __restrict__ X, const float* __restrict__ W,
                        const float* __restrict__ bias, float* __restrict__ Y,
                        int nTiles, int relu) {
  const int tile = blockIdx.x * (blockDim.x >> 5) + (threadIdx.x >> 5);
  if (tile >= nTiles) return;
  v8f acc = V8ZERO;
  acc = wmma_tile16(X + (size_t)tile * 256, W, acc);
  store_tile16(Y + (size_t)tile * 256, acc, bias, relu);
}

// Wide linear: Y[R,Nout] = X[R,16] @ W[16,Nout] + b (for mlp_atom l2, Nout=800)
__global__ void k_lin_wide(const float* __restrict__ X, const float* __restrict__ W,
                           const float* __restrict__ bias, float* __restrict__ Y,
                           int rowTiles, int Nout, int colTiles) {
  const int w = blockIdx.x * (blockDim.x >> 5) + (threadIdx.x >> 5);
  if (w >= rowTiles * colTiles) return;
  const int rt = w / colTiles, ct = w % colTiles;
  const int lane = threadIdx.x & 31;
  const int nm   = lane & 15;
  const int kh   = (lane >> 4) << 1;
  const int ncol = ct * 16 + nm;
  const bool valid = (ncol < Nout);
  v8f acc = V8ZERO;
#pragma unroll
  for (int kk = 0; kk < 4; ++kk) {
    const int k0 = kk * 4 + kh;
    v2f a, b;
    const float* xr = X + ((size_t)rt * 16 + nm) * EMBD + k0;
    a.x = xr[0];
    a.y = xr[1];
    b.x = valid ? W[k0 * Nout + ncol] : 0.0f;
    b.y = valid ? W[(k0 + 1) * Nout + ncol] : 0.0f;
    acc = __builtin_amdgcn_wmma_f32_16x16x4_f32(false, a, false, b,
                                                (short)0, acc, false, false);
  }
  if (valid) {
    const float bv = bias ? bias[ncol] : 0.0f;
    const int mh = (lane >> 4) * 8;
#pragma unroll
    for (int r = 0; r < 8; ++r)
      Y[((size_t)rt * 16 + mh + r) * Nout + ncol] = acc[r] + bv;
  }
}

// ---------------------------------------------------------------------------
// Fused atom-layer node linears: K,Q,V,S = h @ {Wk,Wq,Wv,Wskip}; A-tile shared
// ---------------------------------------------------------------------------
__global__ void k_atom_node4(const float* __restrict__ X,
                             const float* __restrict__ kW, const float* __restrict__ kb,
                             const float* __restrict__ qW, const float* __restrict__ qb,
                             const float* __restrict__ vW, const float* __restrict__ vb,
                             const float* __restrict__ sW,
                             float* __restrict__ K, float* __restrict__ Q,
                             float* __restrict__ V, float* __restrict__ S, int nTiles) {
  const int tile = blockIdx.x * (blockDim.x >> 5) + (threadIdx.x >> 5);
  if (tile >= nTiles) return;
  const float* Xt = X + (size_t)tile * 256;
  v8f a0 = V8ZERO, a1 = V8ZERO, a2 = V8ZERO, a3 = V8ZERO;
  a0 = wmma_tile16(Xt, kW, a0);
  a1 = wmma_tile16(Xt, qW, a1);
  a2 = wmma_tile16(Xt, vW, a2);
  a3 = wmma_tile16(Xt, sW, a3);
  store_tile16(K + (size_t)tile * 256, a0, kb, 0);
  store_tile16(Q + (size_t)tile * 256, a1, qb, 0);
  store_tile16(V + (size_t)tile * 256, a2, vb, 0);
  store_tile16(S + (size_t)tile * 256, a3, nullptr, 0);
}

// Fused bond-layer node linears: P2 = h@W2+b2, P3 = h@W3+b3
__global__ void k_bond_node2(const float* __restrict__ X,
                             const float* __restrict__ w2, const float* __restrict__ b2,
                             const float* __restrict__ w3, const float* __restrict__ b3,
                             float* __restrict__ P2, float* __restrict__ P3, int nTiles) {
  const int tile = blockIdx.x * (blockDim.x >> 5) + (threadIdx.x >> 5);
  if (tile >= nTiles) return;
  const float* Xt = X + (size_t)tile * 256;
  v8f a0 = V8ZERO, a1 = V8ZERO;
  a0 = wmma_tile16(Xt, w2, a0);
  a1 = wmma_tile16(Xt, w3, a1);
  store_tile16(P2 + (size_t)tile * 256, a0, b2, 0);
  store_tile16(P3 + (size_t)tile * 256, a1, b3, 0);
}

// ---------------------------------------------------------------------------
// Fused atom edge: ee = e@We+be (WMMA), gate = sigmoid(K[dst]+Q[src]+ee),
// agg[dst] += gate * V[src]   (D-fragment feeds gate directly, no T2 pass)
// ---------------------------------------------------------------------------
__global__ void k_atom_edge_fused(const float* __restrict__ E,
                                  const float* __restrict__ eW, const float* __restrict__ eb,
                                  const int* __restrict__ src, const int* __restrict__ dst,
                                  const float* __restrict__ K, const float* __restrict__ Q,
                                  const float* __restrict__ V,
                                  float* __restrict__ agg, int nTiles) {
  const int tile = blockIdx.x * (blockDim.x >> 5) + (threadIdx.x >> 5);
  if (tile >= nTiles) return;
  v8f acc = V8ZERO;
  acc = wmma_tile16(E + (size_t)tile * 256, eW, acc);
  const int lane = threadIdx.x & 31;
  const int col  = lane & 15;
  const int mh   = (lane >> 4) * 8;
  const float bv = eb[col];
  const int jb = tile * 16 + mh;
#pragma unroll
  for (int r = 0; r < 8; ++r) {
    const int j = jb + r;
    const int s = src[j], d = dst[j];
    const float g = sigm(K[d * EMBD + col] + Q[s * EMBD + col] + acc[r] + bv);
    __hip_atomic_fetch_add(&agg[d * EMBD + col], g * V[s * EMBD + col],
                           __ATOMIC_RELAXED, __HIP_MEMORY_SCOPE_AGENT);
  }
}

// ---------------------------------------------------------------------------
// Fused bond edge: t = e@W1+b1 + P2[src] + P3[dst]; store t; BN sum/sumsq
// ---------------------------------------------------------------------------
__global__ void k_bond_edge_fused(const float* __restrict__ E,
                                  const float* __restrict__ w1, const float* __restrict__ b1,
                                  const int* __restrict__ src, const int* __restrict__ dst,
                                  const float* __restrict__ P2, const float* __restrict__ P3,
                                  float* __restrict__ T, float* __restrict__ stats,
                                  int nTiles) {
  __shared__ float ssum[EMBD], ssq[EMBD];
  if (threadIdx.x < EMBD) { ssum[threadIdx.x] = 0.f; ssq[threadIdx.x] = 0.f; }
  __syncthreads();
  const int tile = blockIdx.x * (blockDim.x >> 5) + (threadIdx.x >> 5);
  if (tile < nTiles) {
    v8f acc = V8ZERO;
    acc = wmma_tile16(E + (size_t)tile * 256, w1, acc);
    const int lane = threadIdx.x & 31;
    const int col  = lane & 15;
    const int mh   = (lane >> 4) * 8;
    const float bv = b1[col];
    const int jb = tile * 16 + mh;
    float lsum = 0.f, lsq = 0.f;
#pragma unroll
    for (int r = 0; r < 8; ++r) {
      const int j = jb + r;
      const int s = src[j], d = dst[j];
      const float v = acc[r] + bv + P2[s * EMBD + col] + P3[d * EMBD + col];
      T[(size_t)j * EMBD + col] = v;
      lsum += v;
      lsq  += v * v;
    }
    atomicAdd(&ssum[col], lsum);
    atomicAdd(&ssq[col], lsq);
  }
  __syncthreads();
  if (threadIdx.x < EMBD) {
    __hip_atomic_fetch_add(&stats[threadIdx.x], ssum[threadIdx.x],
                           __ATOMIC_RELAXED, __HIP_MEMORY_SCOPE_AGENT);
    __hip_atomic_fetch_add(&stats[EMBD + threadIdx.x], ssq[threadIdx.x],
                           __ATOMIC_RELAXED, __HIP_MEMORY_SCOPE_AGENT);
  }
}

// ---------------------------------------------------------------------------
// Fused z: ea = e@Wa+ba, ed = e@Wd+bd (two WMMAs, shared A), then
// z[batch[src]] += (ea + HB[src] + HC[dst]) * ed
// ---------------------------------------------------------------------------
__global__ void k_z_fused(const float* __restrict__ E,
                          const float* __restrict__ aW, const float* __restrict__ ab,
                          const float* __restrict__ dW, const float* __restrict__ db,
                          const int* __restrict__ src, const int* __restrict__ dst,
                          const int* __restrict__ batch,
                          const float* __restrict__ HB, const float* __restrict__ HC,
                          float* __restrict__ zout, int nTiles) {
  const int tile = blockIdx.x * (blockDim.x >> 5) + (threadIdx.x >> 5);
  if (tile >= nTiles) return;
  const float* Et = E + (size_t)tile * 256;
  v8f aa = V8ZERO, dd = V8ZERO;
  aa = wmma_tile16(Et, aW, aa);
  dd = wmma_tile16(Et, dW, dd);
  const int lane = threadIdx.x & 31;
  const int col  = lane & 15;
  const int mh   = (lane >> 4) * 8;
  const float bva = ab[col], bvd = db[col];
  const int jb = tile * 16 + mh;
#pragma unroll
  for (int r = 0; r < 8; ++r) {
    const int j = jb + r;
    const int s = src[j], d = dst[j];
    const float va = aa[r] + bva + HB[s * EMBD + col] + HC[d * EMBD + col];
    const float vd = dd[r] + bvd;
    __hip_atomic_fetch_add(&zout[batch[s] * EMBD + col], va * vd,
                           __ATOMIC_RELAXED, __HIP_MEMORY_SCOPE_AGENT);
  }
}

// ---------------------------------------------------------------------------
// Fused mlp_bond: hmid = relu(e2@W1+b1) -> LDS transpose -> s = hmid@W2+b2
// (W2 is 16x4; pad B-fragment with zeros, store only 4 cols)
// ---------------------------------------------------------------------------
__global__ void k_mlp_bond_fused(const float* __restrict__ E,
                                 const float* __restrict__ w1, const float* __restrict__ b1,
                                 const float* __restrict__ w2, const float* __restrict__ b2,
                                 float* __restrict__ sout, int nTiles) {
  __shared__ float hmid[4][16][17];   // [wave][row][col], padded
  const int wslot = threadIdx.x >> 5;
  const int tile = blockIdx.x * (blockDim.x >> 5) + wslot;
  if (tile >= nTiles) return;
  v8f acc = V8ZERO;
  acc = wmma_tile16(E + (size_t)tile * 256, w1, acc);
  const int lane = threadIdx.x & 31;
  const int col  = lane & 15;
  const int mh   = (lane >> 4) * 8;
  const int kh   = (lane >> 4) << 1;
  const float bv = b1[col];
#pragma unroll
  for (int r = 0; r < 8; ++r)
    hmid[wslot][mh + r][col] = fmaxf(acc[r] + bv, 0.0f);
  // cross-lane LDS RAW within the wave: wait for our DS stores to land
  asm volatile("s_wait_dscnt 0" ::: "memory");
  v8f acc2 = V8ZERO;
#pragma unroll
  for (int kk = 0; kk < 4; ++kk) {
    const int k0 = kk * 4 + kh;
    v2f a, b;
    a.x = hmid[wslot][col][k0];
    a.y = hmid[wslot][col][k0 + 1];
    b.x = (col < 4) ? w2[k0 * 4 + col] : 0.0f;
    b.y = (col < 4) ? w2[(k0 + 1) * 4 + col] : 0.0f;
    acc2 = __builtin_amdgcn_wmma_f32_16x16x4_f32(false, a, false, b,
                                                 (short)0, acc2, false, false);
  }
  if (col < 4) {
    const float b2v = b2[col];
#pragma unroll
    for (int r = 0; r < 8; ++r)
      sout[(size_t)(tile * 16 + mh + r) * 4 + col] = acc2[r] + b2v;
  }
}

// ---------------------------------------------------------------------------
// Elementwise kernels (float4 vectorized; all counts divisible by 4)
// ---------------------------------------------------------------------------
__global__ void k_zero4(v4f* __restrict__ p, int n4) {
  const int i = blockIdx.x * blockDim.x + threadIdx.x;
  if (i < n4) { v4f z = {0.f, 0.f, 0.f, 0.f}; p[i] = z; }
}

__global__ void k_embed4(const int* __restrict__ idx, const v4f* __restrict__ tab,
                         v4f* __restrict__ out, int n4) {
  const int t = blockIdx.x * blockDim.x + threadIdx.x;
  if (t >= n4) return;
  out[t] = tab[idx[t >> 2] * 4 + (t & 3)];
}

// conv = agg + skip + bias; per-channel sum/sumsq into stats[0..31]
__global__ void k_conv_stats(const float* __restrict__ agg, const float* __restrict__ S,
                             const float* __restrict__ bias, float* __restrict__ conv,
                             float* __restrict__ stats, int n16) {
  __shared__ float ssum[EMBD], ssq[EMBD];
  if (threadIdx.x < EMBD) { ssum[threadIdx.x] = 0.f; ssq[threadIdx.x] = 0.f; }
  __syncthreads();
  const int t = blockIdx.x * blockDim.x + threadIdx.x;
  if (t < n16) {
    const int c = t & 15;
    const float v = agg[t] + S[t] + bias[c];
    conv[t] = v;
    atomicAdd(&ssum[c], v);
    atomicAdd(&ssq[c], v * v);
  }
  __syncthreads();
  if (threadIdx.x < EMBD) {
    __hip_atomic_fetch_add(&stats[threadIdx.x], ssum[threadIdx.x],
                           __ATOMIC_RELAXED, __HIP_MEMORY_SCOPE_AGENT);
    __hip_atomic_fetch_add(&stats[EMBD + threadIdx.x], ssq[threadIdx.x],
                           __ATOMIC_RELAXED, __HIP_MEMORY_SCOPE_AGENT);
  }
}

__global__ void k_bn_final(float* __restrict__ stats, const float* __restrict__ gamma,
                           const float* __restrict__ beta, float invR) {
  const int c = threadIdx.x;
  if (c < EMBD) {
    const float mean = stats[c] * invR;
    const float var  = stats[EMBD + c] * invR - mean * mean;
    const float a    = gamma[c] * rsqrtf(var + 1e-5f);
    stats[32 + c] = a;
    stats[48 + c] = beta[c] - mean * a;
  }
}

// out = base + relu(t*scale[c]+shift[c]); 4 consecutive channels per thread
__global__ void k_add_relu_bn4(const v4f* __restrict__ base, const v4f* __restrict__ t,
                               const float* __restrict__ stats, v4f* __restrict__ out,
                               int n4) {
  const int i = blockIdx.x * blockDim.x + threadIdx.x;
  if (i >= n4) return;
  const int c0 = (i & 3) * 4;
  const v4f b = base[i], tv = t[i];
  v4f o;
#pragma unroll
  for (int q = 0; q < 4; ++q)
    o[q] = b[q] + fmaxf(tv[q] * stats[32 + c0 + q] + stats[48 + c0 + q], 0.0f);
  out[i] = o;
}

// e2[j] = zb[batch[fsrc[j]]] (4 channels per thread)
__global__ void k_bcast_e2_4(const int* __restrict__ fsrc, const int* __restrict__ batch,
                             const v4f* __restrict__ zb, v4f* __restrict__ e2, int n4) {
  const int t = blockIdx.x * blockDim.x + threadIdx.x;
  if (t >= n4) return;
  e2[t] = zb[batch[fsrc[t >> 2]] * 4 + (t & 3)];
}

// ---------------------------------------------------------------------------
// Host orchestration. Param leaves in jax pytree order (sorted dict keys):
//   0:a.W 1:a.b | 2..49:adec[l]x12 | 50:aemb | 51..98:aenc[l]x12 | 99:b.W
//   100:b.b | 101..132:bdec[l]x8 | 133:bemb | 134..165:benc[l]x8 | 166:c.W
//   167:c.b | 168:d.W 169:d.b | 170..173:mlp_atom | 174..177:mlp_bond |
//   178:u.W 179:u.b | 180:x 181:edge_index 182:edge_attr 183:batch
//   184:fc_edge_index
// ---------------------------------------------------------------------------
extern "C" void kernel_launch(void* const* d_in, const int* in_sizes, int n_in,
                              void* d_out, int out_size, void* d_ws, size_t ws_size,
                              hipStream_t stream) {
  const float* const* P = (const float* const*)d_in;
  const int* x     = (const int*)d_in[180];
  const int* ei    = (const int*)d_in[181];
  const int* eattr = (const int*)d_in[182];
  const int* batch = (const int*)d_in[183];
  const int* fei   = (const int*)d_in[184];
  const int* esrc = ei,  *edst = ei + N_EDGES;
  const int* fsrc = fei, *fdst = fei + FC_E;

  float* ws = (float*)d_ws;
  const size_t NH   = (size_t)N_NODES * EMBD;   // 524288
  const size_t FE16 = (size_t)FC_E * EMBD;      // 16252928
  float* H    = ws + 0 * NH;
  float* HN   = ws + 1 * NH;
  float* Kp   = ws + 2 * NH;
  float* Qp   = ws + 3 * NH;
  float* Vp   = ws + 4 * NH;
  float* Sp   = ws + 5 * NH;
  float* AGG  = ws + 6 * NH;
  float* CONV = ws + 7 * NH;
  float* P2   = ws + 8 * NH;
  float* P3   = ws + 9 * NH;
  float* STATS = ws + 10 * NH;                  // 64 floats
  float* ZB   = STATS + 64;                     // [B,16]
  float* E2   = ZB + (size_t)NGRAPH * EMBD;     // [FC_E,16]; encoder e in front
  float* T2   = E2 + FE16;                      // [FC_E,16] bond-layer t

  float* out     = (float*)d_out;
  float* out_boa = out;                 // [1024, 800]
  float* out_z   = out + 819200;        // [1024, 16]
  float* out_s   = out + 835584;        // [FC_E, 4]

  const int TILES_N  = N_NODES / 16;    // 2048
  const int TILES_E  = N_EDGES / 16;    // 4096
  const int TILES_FE = FC_E / 16;       // 63488
  const int TILES_B  = NGRAPH / 16;     // 64

  auto lgrid = [](int tiles) { return dim3((unsigned)((tiles + 3) / 4)); };
  auto egrid = [](int n) { return dim3((unsigned)((n + 255) / 256)); };

  auto run_phase = [&](const float* const* Pp, int abase, int bbase,
                       const int* src, const int* dst, int ecnt, float* Ebuf) {
    const int tilesE = ecnt / 16;
    const int e16 = ecnt * EMBD;
    for (int l = 0; l < NUM_LAYERS; ++l) {
      const float* A_bias = Pp[abase + 12 * l + 0];
      const float* A_bnb  = Pp[abase + 12 * l + 1];
      const float* A_bng  = Pp[abase + 12 * l + 2];
      const float* A_eW   = Pp[abase + 12 * l + 3];
      const float* A_eb   = Pp[abase + 12 * l + 4];
      const float* A_kW   = Pp[abase + 12 * l + 5];
      const float* A_kb   = Pp[abase + 12 * l + 6];
      const float* A_qW   = Pp[abase + 12 * l + 7];
      const float* A_qb   = Pp[abase + 12 * l + 8];
      const float* A_sW   = Pp[abase + 12 * l + 9];
      const float* A_vW   = Pp[abase + 12 * l + 10];
      const float* A_vb   = Pp[abase + 12 * l + 11];
      const float* B_bnb  = Pp[bbase + 8 * l + 0];
      const float* B_bng  = Pp[bbase + 8 * l + 1];
      const float* B_v1W  = Pp[bbase + 8 * l + 2];
      const float* B_v1b  = Pp[bbase + 8 * l + 3];
      const float* B_v2W  = Pp[bbase + 8 * l + 4];
      const float* B_v2b  = Pp[bbase + 8 * l + 5];
      const float* B_v3W  = Pp[bbase + 8 * l + 6];
      const float* B_v3b  = Pp[bbase + 8 * l + 7];

      // ---- atom layer ----
      k_atom_node4<<<lgrid(TILES_N), 128, 0, stream>>>(
          H, A_kW, A_kb, A_qW, A_qb, A_vW, A_vb, A_sW, Kp, Qp, Vp, Sp, TILES_N);
      k_zero4<<<egrid((int)NH / 4), 256, 0, stream>>>((v4f*)AGG, (int)NH / 4);
      k_zero4<<<1, 16, 0, stream>>>((v4f*)STATS, 16);
      k_atom_edge_fused<<<lgrid(tilesE), 128, 0, stream>>>(
          Ebuf, A_eW, A_eb, src, dst, Kp, Qp, Vp, AGG, tilesE);
      k_conv_stats<<<egrid((int)NH), 256, 0, stream>>>(AGG, Sp, A_bias, CONV, STATS, (int)NH);
      k_bn_final<<<1, 16, 0, stream>>>(STATS, A_bng, A_bnb, 1.0f / (float)N_NODES);
      k_add_relu_bn4<<<egrid((int)NH / 4), 256, 0, stream>>>(
          (const v4f*)H, (const v4f*)CONV, STATS, (v4f*)HN, (int)NH / 4);

      // ---- bond layer (pre-update H) ----
      k_bond_node2<<<lgrid(TILES_N), 128, 0, stream>>>(
          H, B_v2W, B_v2b, B_v3W, B_v3b, P2, P3, TILES_N);
      k_zero4<<<1, 16, 0, stream>>>((v4f*)STATS, 16);
      k_bond_edge_fused<<<lgrid(tilesE), 128, 0, stream>>>(
          Ebuf, B_v1W, B_v1b, src, dst, P2, P3, T2, STATS, tilesE);
      k_bn_final<<<1, 16, 0, stream>>>(STATS, B_bng, B_bnb, 1.0f / (float)ecnt);
      k_add_relu_bn4<<<egrid(e16 / 4), 256, 0, stream>>>(
          (const v4f*)Ebuf, (const v4f*)T2, STATS, (v4f*)Ebuf, e16 / 4);

      float* tmp = H; H = HN; HN = tmp;  // h = h_new
    }
  };

  // ---- embeddings ----
  k_embed4<<<egrid((int)NH / 4), 256, 0, stream>>>(x, (const v4f*)P[50], (v4f*)H, (int)NH / 4);
  k_embed4<<<egrid(N_EDGES * 4), 256, 0, stream>>>(eattr, (const v4f*)P[133], (v4f*)E2,
                                                   N_EDGES * 4);

  // ---- encoder: aenc base 51, benc base 134 ----
  run_phase(P, 51, 134, esrc, edst, N_EDGES, E2);

  // ---- z = segsum((e@a + h[src]@b + h[dst]@c) * (e@d)) ----
  k_lin16<<<lgrid(TILES_N), 128, 0, stream>>>(H, P[99],  P[100], Kp, TILES_N, 0); // hb
  k_lin16<<<lgrid(TILES_N), 128, 0, stream>>>(H, P[166], P[167], Qp, TILES_N, 0); // hc
  k_zero4<<<egrid(NGRAPH * 4), 256, 0, stream>>>((v4f*)out_z, NGRAPH * 4);
  k_z_fused<<<lgrid(TILES_E), 128, 0, stream>>>(E2, P[0], P[1], P[168], P[169],
                                                esrc, edst, batch, Kp, Qp, out_z, TILES_E);

  // ---- boa = mlp2(z, mlp_atom) ----
  k_lin16<<<lgrid(TILES_B), 128, 0, stream>>>(out_z, P[170], P[171], ZB, TILES_B, 1);
  k_lin_wide<<<lgrid(TILES_B * 50), 128, 0, stream>>>(ZB, P[172], P[173], out_boa,
                                                      TILES_B, 800, 50);

  // ---- decoder init: e2 = (z@u+ub)[graph], h2 = aemb[x] ----
  k_lin16<<<lgrid(TILES_B), 128, 0, stream>>>(out_z, P[178], P[179], ZB, TILES_B, 0);
  k_bcast_e2_4<<<egrid(FC_E * 4), 256, 0, stream>>>(fsrc, batch, (const v4f*)ZB,
                                                    (v4f*)E2, FC_E * 4);
  k_embed4<<<egrid((int)NH / 4), 256, 0, stream>>>(x, (const v4f*)P[50], (v4f*)H, (int)NH / 4);

  // ---- decoder: adec base 2, bdec base 101 ----
  run_phase(P, 2, 101, fsrc, fdst, FC_E, E2);

  // ---- s = mlp2(e2, mlp_bond), fused through LDS ----
  k_mlp_bond_fused<<<lgrid(TILES_FE), 128, 0, stream>>>(E2, P[174], P[175],
                                                        P[176], P[177], out_s, TILES_FE);

  (void)in_sizes; (void)n_in; (void)out_size; (void)ws_size;
}